// QuantumAttentionBlock_65481071401834
// MI455X (gfx1250) — compile-verified
//
#include <hip/hip_runtime.h>
#include <hip/hip_bf16.h>
#include <stdint.h>

// ---------------------------------------------------------------------------
// QuantumAttentionBlock: out = cos(x) * cos(theta[e % 64])
// Memory-bound streaming map (268 MB @ 23.3 TB/s ~ 11.5 us floor).
// CDNA5 path: TDM tensor_load_to_lds double-buffered HBM->LDS staging,
// s_wait_tensorcnt, ds_load_b128 consume, non-temporal global_store_b128.
// ---------------------------------------------------------------------------

typedef __attribute__((ext_vector_type(4))) float        v4f;
typedef __attribute__((ext_vector_type(4))) unsigned int v4u;
typedef __attribute__((ext_vector_type(8))) int          v8i;
typedef __attribute__((ext_vector_type(4))) int          v4i;

#define TILE_ELEMS 8192                    // 32 KB per tile
#define THREADS    256                     // 8 wave32
#define CHUNKS     (TILE_ELEMS / (THREADS * 4))   // 8 x float4 per thread per tile

// Build a 1-D TDM descriptor (D#) and issue TENSOR_LOAD_TO_LDS.
// Group0: count=1 | lds_addr | global_addr(57b) | type=2
// Group1: data_size=4B, tensor_dim0=tile_dim0=nelem, tensor_dim1=tile_dim1=1
__device__ __forceinline__ void tdm_load_tile_f32(const float* gsrc,
                                                  unsigned lds_byte_off,
                                                  unsigned nelem) {
  unsigned long long ga = (unsigned long long)(uintptr_t)gsrc;
  v4u g0;
  g0.x = 1u;                                              // count=1, user mode
  g0.y = lds_byte_off;                                    // lds_addr [63:32]
  g0.z = (unsigned)ga;                                    // global_addr lo32
  g0.w = (unsigned)((ga >> 32) & 0x01FFFFFFull)           // global_addr[56:32]
         | (2u << 30);                                    // type = 2 (image)
  v8i g1;
  g1[0] = 0x00020000;                                     // wg_mask=0, data_size=2 (4B)
  g1[1] = (int)((nelem & 0xFFFFu) << 16);                 // tensor_dim0[15:0]
  g1[2] = (int)((nelem >> 16) & 0xFFFFu) | (1 << 16);     // tensor_dim0[31:16], tensor_dim1=1
  g1[3] = (int)((nelem & 0xFFFFu) << 16);                 // tile_dim0 = nelem
  g1[4] = 1;                                              // tile_dim1 = 1, tile_dim2 = 0
  g1[5] = (int)nelem;                                     // tensor_dim0_stride lo32
  g1[6] = 0;                                              // stride hi, dim1_stride lo
  g1[7] = 0;
  v4i z4 = {0, 0, 0, 0};                                  // groups 2/3 unused (<=2D)
  v8i z8 = {0, 0, 0, 0, 0, 0, 0, 0};                      // unused 5th operand
  __builtin_amdgcn_tensor_load_to_lds(g0, g1, z4, z4, z8, /*cpol=*/0);
}

__global__ void __launch_bounds__(THREADS)
qab_tdm_kernel(const float* __restrict__ x, const float* __restrict__ theta,
               float* __restrict__ out, int tilesPerBlock) {
  __shared__ __align__(16) float buf[2][TILE_ELEMS];

  const int tid       = threadIdx.x;
  const int firstTile = blockIdx.x * tilesPerBlock;

  // Loop-invariant cos(theta) vector: tile size (8192) and chunk stride (1024)
  // are multiples of HD=64, so each thread's 4 head-dim slots never change.
  const int h = (tid * 4) & 63;
  v4f th = *reinterpret_cast<const v4f*>(theta + h);
  v4f c;
  c.x = __cosf(th.x); c.y = __cosf(th.y);
  c.z = __cosf(th.z); c.w = __cosf(th.w);

  const bool wave0 = (tid < 32);
  // Generic LDS pointer = {shared aperture, 32-bit LDS byte offset}; truncate.
  const unsigned lds0 = (unsigned)(uintptr_t)&buf[0][0];
  const unsigned lds1 = (unsigned)(uintptr_t)&buf[1][0];

  if (wave0) {
    tdm_load_tile_f32(x + (size_t)firstTile * TILE_ELEMS, lds0, TILE_ELEMS);
  }

  for (int i = 0; i < tilesPerBlock; ++i) {
    const int cur = i & 1;
    if (wave0) {
      if (i + 1 < tilesPerBlock) {
        // Prefetch next tile into the other buffer, then wait for current.
        tdm_load_tile_f32(x + (size_t)(firstTile + i + 1) * TILE_ELEMS,
                          cur ? lds0 : lds1, TILE_ELEMS);
        __builtin_amdgcn_s_wait_tensorcnt(1);   // oldest (tile i) complete
      } else {
        __builtin_amdgcn_s_wait_tensorcnt(0);   // drain
      }
    }
    __syncthreads();                            // tile i visible to all waves

    const float* src  = buf[cur];
    const size_t base = (size_t)(firstTile + i) * TILE_ELEMS;
#pragma unroll
    for (int j = 0; j < CHUNKS; ++j) {
      const int local = tid * 4 + j * (THREADS * 4);
      v4f v = *reinterpret_cast<const v4f*>(src + local);   // ds_load_b128
      v4f r;
      r.x = __cosf(v.x) * c.x;
      r.y = __cosf(v.y) * c.y;
      r.z = __cosf(v.z) * c.z;
      r.w = __cosf(v.w) * c.w;
      // Write-once output: non-temporal 128-bit store, skip cache residency.
      __builtin_nontemporal_store(
          r, reinterpret_cast<v4f*>(out + base + local));
    }
    __syncthreads();   // all readers done with buf[cur] before it is re-filled
  }
}

// Fallback for any tail not covered by whole tiles (unused for this shape).
__global__ void qab_tail_kernel(const float* __restrict__ x,
                                const float* __restrict__ theta,
                                float* __restrict__ out,
                                int start, int n) {
  int i = start + blockIdx.x * blockDim.x + threadIdx.x;
  if (i < n) {
    out[i] = __cosf(x[i]) * __cosf(theta[i & 63]);
  }
}

extern "C" void kernel_launch(void* const* d_in, const int* in_sizes, int n_in,
                              void* d_out, int out_size, void* d_ws, size_t ws_size,
                              hipStream_t stream) {
  const float* x     = (const float*)d_in[0];
  const float* theta = (const float*)d_in[1];
  float*       out   = (float*)d_out;

  const int n      = in_sizes[0];            // 33,554,432 for B,S,E = 8,4096,1024
  const int ntiles = n / TILE_ELEMS;         // 4096
  int tpb = 4;                               // tiles per block -> 1024 blocks
  if (tpb > ntiles || (ntiles % tpb) != 0) tpb = 1;
  const int blocks = (ntiles > 0) ? (ntiles / tpb) : 0;

  if (blocks > 0) {
    qab_tdm_kernel<<<blocks, THREADS, 0, stream>>>(x, theta, out, tpb);
  }
  const int covered = ntiles * TILE_ELEMS;
  const int tail    = n - covered;
  if (tail > 0) {
    qab_tail_kernel<<<(tail + THREADS - 1) / THREADS, THREADS, 0, stream>>>(
        x, theta, out, covered, n);
  }
}